// DetrLoss_20495583936553
// MI455X (gfx1250) — compile-verified
//
#include <hip/hip_runtime.h>
#include <stdint.h>

#define Q   900
#define TI  64          // tile extent along i (gt index, contiguous in pred_cls)
#define TJ  64          // tile extent along j (pred index)
#define TIP 65          // padded LDS row stride (floats) -> conflict-free transpose reads

typedef unsigned int uint32x4 __attribute__((ext_vector_type(4)));
typedef int          int32x8  __attribute__((ext_vector_type(8)));
typedef int          int32x4  __attribute__((ext_vector_type(4)));

__global__ __launch_bounds__(256)
void detr_cost_kernel(const float* __restrict__ pred_boxes,
                      const float* __restrict__ pred_cls,
                      const float* __restrict__ gt_boxes,
                      const int*   __restrict__ gt_valid,
                      float* __restrict__ out)
{
    __shared__ float  s_cls[TJ * TIP];   // pred_cls tile, [jj][ii], row stride TIP
    __shared__ float4 s_pb[TJ];
    __shared__ float4 s_gt[TI];
    __shared__ int    s_val[TI];

    const int b   = blockIdx.z;
    const int i0  = blockIdx.y * TI;
    const int j0  = blockIdx.x * TJ;
    const int tid = threadIdx.x;

    if (tid < 32) {
        // ---- wave 0: TDM async 2D tile load pred_cls[b, j0:j0+TJ, i0:i0+TI] -> LDS ----
        unsigned long long gaddr = (unsigned long long)(size_t)pred_cls
            + 4ull * ((unsigned long long)b * Q * Q
                    + (unsigned long long)(unsigned)j0 * Q
                    + (unsigned)i0);
        unsigned lds   = (unsigned)(size_t)(void*)s_cls;   // low 32 bits of flat LDS addr = LDS offset
        unsigned tdim0 = (unsigned)(Q - i0);               // remaining extent -> OOB zero-fill at edges
        unsigned tdim1 = (unsigned)(Q - j0);

        uint32x4 g0;
        g0[0] = 1u;                                        // count = 1 valid descriptor
        g0[1] = lds;                                       // lds_addr (bytes)
        g0[2] = (unsigned)(gaddr & 0xFFFFFFFFull);         // global_addr[31:0]
        g0[3] = ((unsigned)(gaddr >> 32) & 0x01FFFFFFu)    // global_addr[56:32]
              | (2u << 30);                                // type = 2 ("image")

        int32x8 g1;
        g1[0] = (int)((2u << 16)      // data_size = 4 bytes
                    | (1u << 20)      // pad_enable
                    | (5u << 22));    // pad_interval: 64 dwords; pad_amount field 0 => 1 dword
        g1[1] = (int)((tdim0 & 0xFFFFu) << 16);                       // tensor_dim0[15:0]
        g1[2] = (int)((tdim0 >> 16) | ((tdim1 & 0xFFFFu) << 16));     // tensor_dim0[31:16] | tensor_dim1[15:0]
        g1[3] = (int)((tdim1 >> 16) | ((unsigned)TI << 16));          // tensor_dim1[31:16] | tile_dim0
        g1[4] = (int)TJ;                                              // tile_dim1 (tile_dim2 = 0)
        g1[5] = (int)Q;                                               // tensor_dim0_stride (low 32)
        g1[6] = 0;
        g1[7] = 0;

        int32x4 gz; gz[0] = 0; gz[1] = 0; gz[2] = 0; gz[3] = 0;
#if defined(__clang_major__) && (__clang_major__ >= 23)
        int32x8 gz8;
        gz8[0]=0; gz8[1]=0; gz8[2]=0; gz8[3]=0; gz8[4]=0; gz8[5]=0; gz8[6]=0; gz8[7]=0;
        __builtin_amdgcn_tensor_load_to_lds(g0, g1, gz, gz, gz8, 0);
#else
        __builtin_amdgcn_tensor_load_to_lds(g0, g1, gz, gz, 0);
#endif
        __builtin_amdgcn_s_wait_tensorcnt(0);
    } else if (tid < 96) {
        // ---- waves 1-2: gt boxes + validity for this i tile ----
        int t  = tid - 32;
        int gi = i0 + t;
        int gc = gi < Q ? gi : (Q - 1);
        const float4* gb4 = (const float4*)gt_boxes;
        s_gt[t]  = gb4[(size_t)b * Q + gc];
        s_val[t] = (gi < Q) ? gt_valid[(size_t)b * Q + gc] : 0;
    } else if (tid < 160) {
        // ---- waves 3-4: pred boxes for this j tile ----
        int t  = tid - 96;
        int pj = j0 + t;
        int pc = pj < Q ? pj : (Q - 1);
        const float4* pb4 = (const float4*)pred_boxes;
        s_pb[t] = pb4[(size_t)b * Q + pc];
    }
    __syncthreads();

    // ---- compute: each thread owns column jj, 16 rows ii ----
    const int   jj  = tid & 63;
    const int   iw  = tid >> 6;        // 0..3, uniform within each wave
    const int   j   = j0 + jj;
    const bool  jok = (j < Q);
    const float4 pb = s_pb[jj];
    const float area_p = (pb.z - pb.x) * (pb.w - pb.y);

#pragma unroll
    for (int k = 0; k < 16; ++k) {
        const int ii = iw * 16 + k;
        const int i  = i0 + ii;

        const float4 g = s_gt[ii];                       // broadcast read
        const float area_g = (g.z - g.x) * (g.w - g.y);

        // intersection
        const float x1i = fmaxf(pb.x, g.x);
        const float y1i = fmaxf(pb.y, g.y);
        const float x2i = fminf(pb.z, g.z);
        const float y2i = fminf(pb.w, g.w);
        const float inter = fmaxf(x2i - x1i, 0.0f) * fmaxf(y2i - y1i, 0.0f);
        const float uni   = area_p + area_g - inter;
        const float iou   = inter * __builtin_amdgcn_rcpf(uni + 1e-7f);

        // enclosing box
        const float xc1 = fminf(pb.x, g.x);
        const float yc1 = fminf(pb.y, g.y);
        const float xc2 = fmaxf(pb.z, g.z);
        const float yc2 = fmaxf(pb.w, g.w);
        const float area_c = (xc2 - xc1) * (yc2 - yc1);
        const float giou = iou - (area_c - uni) * __builtin_amdgcn_rcpf(area_c + 1e-7f);

        const float l1 = 0.25f * (fabsf(pb.x - g.x) + fabsf(pb.y - g.y)
                                + fabsf(pb.z - g.z) + fabsf(pb.w - g.w));

        const float cls = s_cls[jj * TIP + ii];          // transposed, bank-conflict-free
        float cost = (l1 + (1.0f - giou)) - cls;
        cost = s_val[ii] ? cost : 0.0f;

        if (jok && i < Q)
            out[((size_t)b * Q + i) * Q + j] = cost;
    }
}

extern "C" void kernel_launch(void* const* d_in, const int* in_sizes, int n_in,
                              void* d_out, int out_size, void* d_ws, size_t ws_size,
                              hipStream_t stream) {
    const float* pred_boxes = (const float*)d_in[0];
    const float* pred_cls   = (const float*)d_in[1];
    const float* gt_boxes   = (const float*)d_in[2];
    const int*   gt_valid   = (const int*)d_in[3];
    float*       out        = (float*)d_out;

    const int B = in_sizes[0] / (Q * 4);                 // 32
    dim3 grid((Q + TJ - 1) / TJ, (Q + TI - 1) / TI, B);  // (15, 15, 32)
    detr_cost_kernel<<<grid, dim3(256), 0, stream>>>(pred_boxes, pred_cls,
                                                     gt_boxes, gt_valid, out);
    (void)n_in; (void)out_size; (void)d_ws; (void)ws_size;
}